// Render_Mask_13889924235541
// MI455X (gfx1250) — compile-verified
//
#include <hip/hip_runtime.h>
#include <math.h>

typedef float v2f __attribute__((ext_vector_type(2)));
typedef float v8f __attribute__((ext_vector_type(8)));

#define SIGMA_INV 10000.0f
#define BLUR_C    9.2102403669758494e-4f   // log(1/1e-4 - 1) * 1e-4
#define LOG2E     1.4426950408889634f
#define CHUNK     128                       // faces staged in LDS per block pass

// ---------------------------------------------------------------------------
// Kernel 1: per-face preprocessing (unchanged from round 1).
// 9 affine-coefficient float4s over pixel features (px, py, r2, 1) per face:
//   j=0..2 : edge functions w0,w1,w2
//   j=3..5 : s_k = ap.ab for edges (v0,v1),(v1,v2),(v2,v0)
//   j=6..8 : q_k = |p-a|^2 for the same edges
// fc[2f] = (area, invL0..2), fc[2f+1] = (L0..2, zok). Padding faces: zok=0.
// ---------------------------------------------------------------------------
__global__ void rm_prep_kernel(const float* __restrict__ verts,
                               const int*   __restrict__ faces,
                               const float* __restrict__ focal,
                               const float* __restrict__ princ,
                               const int* __restrict__ pH,
                               const int* __restrict__ pW,
                               int F, int F16,
                               float4* __restrict__ coef,   // [9][F16]
                               float4* __restrict__ fc)     // [2*F16]
{
    int f = blockIdx.x * blockDim.x + threadIdx.x;
    if (f >= F16) return;
    if (f >= F) {
        float4 z4 = make_float4(0.f, 0.f, 0.f, 0.f);
        #pragma unroll
        for (int j = 0; j < 9; ++j) coef[j * F16 + f] = z4;
        fc[2 * f + 0] = z4;
        fc[2 * f + 1] = z4;
        return;
    }
    const int H = *pH, W = *pW;
    const float s    = (float)(H < W ? H : W);
    const float fclx = focal[0] * 2.0f / s;
    const float fcly = focal[1] * 2.0f / s;
    const float prx  = -(princ[0] - (float)(W / 2)) * 2.0f / (float)W;
    const float pry  = -(princ[1] - (float)(H / 2)) * 2.0f / (float)H;

    int i0 = faces[3 * f + 0], i1 = faces[3 * f + 1], i2 = faces[3 * f + 2];
    float x0 = verts[3 * i0], y0 = verts[3 * i0 + 1], z0 = verts[3 * i0 + 2];
    float x1 = verts[3 * i1], y1 = verts[3 * i1 + 1], z1 = verts[3 * i1 + 2];
    float x2 = verts[3 * i2], y2 = verts[3 * i2 + 1], z2 = verts[3 * i2 + 2];

    float X0 = -fclx * x0 / z0 + prx, Y0 = -fcly * y0 / z0 + pry;
    float X1 = -fclx * x1 / z1 + prx, Y1 = -fcly * y1 / z1 + pry;
    float X2 = -fclx * x2 / z2 + prx, Y2 = -fcly * y2 / z2 + pry;

    float area = (X1 - X0) * (Y2 - Y0) - (Y1 - Y0) * (X2 - X0);
    float zok  = (fminf(z0, fminf(z1, z2)) > 0.01f) ? 1.0f : 0.0f;

    coef[0 * F16 + f] = make_float4(-(Y2 - Y1), (X2 - X1), 0.f, X1 * Y2 - X2 * Y1); // w0
    coef[1 * F16 + f] = make_float4(-(Y0 - Y2), (X0 - X2), 0.f, X2 * Y0 - X0 * Y2); // w1
    coef[2 * F16 + f] = make_float4(-(Y1 - Y0), (X1 - X0), 0.f, X0 * Y1 - X1 * Y0); // w2

    float ax[3] = {X0, X1, X2}, ay[3] = {Y0, Y1, Y2};
    float bx[3] = {X1, X2, X0}, by[3] = {Y1, Y2, Y0};
    float L[3], invL[3];
    #pragma unroll
    for (int k = 0; k < 3; ++k) {
        float abx = bx[k] - ax[k], aby = by[k] - ay[k];
        L[k]    = abx * abx + aby * aby;
        invL[k] = 1.0f / fmaxf(L[k], 1e-12f);
        coef[(3 + k) * F16 + f] =
            make_float4(abx, aby, 0.f, -(ax[k] * abx + ay[k] * aby));
        coef[(6 + k) * F16 + f] =
            make_float4(-2.0f * ax[k], -2.0f * ay[k], 1.0f,
                        ax[k] * ax[k] + ay[k] * ay[k]);
    }
    fc[2 * f + 0] = make_float4(area, invL[0], invL[1], invL[2]);
    fc[2 * f + 1] = make_float4(L[0], L[1], L[2], zok);
}

// ---------------------------------------------------------------------------
// Kernel 2: rasterizer. One wave32 per 16-pixel tile; 8 waves per block share
// face data staged through LDS in 128-face chunks. Per 16-face group:
// 9 x V_WMMA_F32_16X16X4_F32 (A = pixel features, B = face coefficients from
// LDS) followed by a branchless clamp/min/sigmoid epilogue.
// ---------------------------------------------------------------------------
__global__ void __launch_bounds__(256)
rm_raster_kernel(const float4* __restrict__ coef,
                 const float4* __restrict__ fc,
                 const int* __restrict__ pH,
                 const int* __restrict__ pW,
                 int F16, int npix,
                 float* __restrict__ out)
{
    __shared__ float4 sBuf[11 * CHUNK];   // [0..8]*CHUNK coef, 9*CHUNK fcA, 10*CHUNK fcB

    const int lane = threadIdx.x & 31;
    const int wave = threadIdx.x >> 5;
    const int tile = blockIdx.x * (blockDim.x >> 5) + wave;
    const int base = tile * 16;

    const int   H  = *pH, W = *pW;
    const float Wf = (float)W, Hf = (float)H;

    // A-matrix fragment: pixel features for row M = lane & 15 (clamped decode
    // for out-of-range tiles; their results are never stored).
    int pm = base + (lane & 15);
    if (pm >= npix) pm = npix - 1;
    const int   ix  = pm % W, iy = pm / W;
    const float px  = 1.0f - (2.0f * (float)ix + 1.0f) / Wf;
    const float py  = 1.0f - (2.0f * (float)iy + 1.0f) / Hf;
    const float r2  = px * px + py * py;
    const bool  hi  = lane >= 16;
    v2f A;
    A.x = hi ? r2 : px;
    A.y = hi ? 1.0f : py;

    float acc[8];
    #pragma unroll
    for (int v = 0; v < 8; ++v) acc[v] = 1.0f;

    const int fl = lane & 15;

    for (int c0 = 0; c0 < F16; c0 += CHUNK) {
        __syncthreads();
        // Cooperative global -> LDS stage of this chunk's face data.
        for (int e = threadIdx.x; e < 11 * CHUNK; e += 256) {
            int j = e / CHUNK, i = e % CHUNK;
            int fi = c0 + i;
            float4 val = make_float4(0.f, 0.f, 0.f, 0.f);
            if (fi < F16) {
                if (j < 9)       val = coef[j * F16 + fi];
                else if (j == 9) val = fc[2 * fi + 0];
                else             val = fc[2 * fi + 1];
            }
            sBuf[e] = val;
        }
        __syncthreads();

        for (int g = 0; g < CHUNK; g += 16) {
            const int fi = g + fl;                 // face column within chunk
            const float4 ca = sBuf[9  * CHUNK + fi];
            const float4 cb = sBuf[10 * CHUNK + fi];

            v8f D[9];
            #pragma unroll
            for (int j = 0; j < 9; ++j) {
                float4 c = sBuf[j * CHUNK + fi];
                v2f B;
                B.x = hi ? c.z : c.x;
                B.y = hi ? c.w : c.y;
                v8f C = {0.f, 0.f, 0.f, 0.f, 0.f, 0.f, 0.f, 0.f};
                D[j] = __builtin_amdgcn_wmma_f32_16x16x4_f32(
                    false, A, false, B, (short)0, C, false, false);
            }

            const float area = ca.x;
            const float iL0 = ca.y, iL1 = ca.z, iL2 = ca.w;
            const float L0  = cb.x, L1  = cb.y, L2  = cb.z;
            const bool  zok = cb.w != 0.0f;

            #pragma unroll
            for (int v = 0; v < 8; ++v) {
                float w0 = D[0][v], w1 = D[1][v], w2 = D[2][v];
                float s0 = D[3][v], s1 = D[4][v], s2 = D[5][v];
                float q0 = D[6][v], q1 = D[7][v], q2 = D[8][v];

                // literal w*area >= 0 (area==0 -> inside, matching reference)
                bool inside = (w0 * area >= 0.0f) && (w1 * area >= 0.0f) &&
                              (w2 * area >= 0.0f);

                float t0 = fminf(fmaxf(s0 * iL0, 0.0f), 1.0f);
                float t1 = fminf(fmaxf(s1 * iL1, 0.0f), 1.0f);
                float t2 = fminf(fmaxf(s2 * iL2, 0.0f), 1.0f);
                float d0 = fmaf(t0, fmaf(t0, L0, -2.0f * s0), q0);
                float d1 = fmaf(t1, fmaf(t1, L1, -2.0f * s1), q1);
                float d2 = fmaf(t2, fmaf(t2, L2, -2.0f * s2), q2);
                float mind2 = fmaxf(fminf(d0, fminf(d1, d2)), 0.0f);

                float sd    = inside ? -mind2 : mind2;
                bool  valid = (sd <= BLUR_C) && zok;

                // branchless sigmoid(-sd/SIGMA) = 1/(1 + 2^(sd*1e4*log2e))
                float e  = __builtin_amdgcn_exp2f(sd * (SIGMA_INV * LOG2E));
                float pr = __builtin_amdgcn_rcpf(1.0f + e);
                float prob = valid ? pr : 0.0f;
                acc[v] *= (1.0f - prob);
            }
        }
    }

    // Product-reduce across each 16-lane half (masks 1,2,4,8 stay in-half).
    #pragma unroll
    for (int v = 0; v < 8; ++v) {
        float a = acc[v];
        a *= __shfl_xor(a, 1, 32);
        a *= __shfl_xor(a, 2, 32);
        a *= __shfl_xor(a, 4, 32);
        a *= __shfl_xor(a, 8, 32);
        acc[v] = a;
    }

    if ((lane & 15) == 0) {
        const int moff = hi ? 8 : 0;        // lanes>=16 own pixel rows 8..15
        #pragma unroll
        for (int v = 0; v < 8; ++v) {
            int p = base + moff + v;
            if (p < npix) out[p] = 1.0f - acc[v];
        }
    }
}

extern "C" void kernel_launch(void* const* d_in, const int* in_sizes, int n_in,
                              void* d_out, int out_size, void* d_ws, size_t ws_size,
                              hipStream_t stream) {
    const float* verts = (const float*)d_in[0];
    const int*   faces = (const int*)d_in[1];
    const float* focal = (const float*)d_in[2];
    const float* princ = (const float*)d_in[3];
    const int*   pH    = (const int*)d_in[4];
    const int*   pW    = (const int*)d_in[5];

    const int F   = in_sizes[1] / 3;
    const int F16 = ((F + 15) / 16) * 16;

    float4* coef = (float4*)d_ws;          // 9 * F16 float4
    float4* fc   = coef + 9 * F16;         // 2 * F16 float4  (~273 KB total)

    const int npix = out_size;             // H*W (B=1, single channel)

    {
        int threads = 256;
        int blocks  = (F16 + threads - 1) / threads;
        rm_prep_kernel<<<blocks, threads, 0, stream>>>(
            verts, faces, focal, princ, pH, pW, F, F16, coef, fc);
    }
    {
        const int tiles   = (npix + 15) / 16;   // one wave32 per 16 pixels
        const int wavesPB = 8;                  // 256-thread blocks
        int blocks = (tiles + wavesPB - 1) / wavesPB;
        rm_raster_kernel<<<blocks, wavesPB * 32, 0, stream>>>(
            coef, fc, pH, pW, F16, npix, (float*)d_out);
    }
}